// ColumnStochasticGraphConvolution_28759101014306
// MI455X (gfx1250) — compile-verified
//
#include <hip/hip_runtime.h>

typedef __attribute__((ext_vector_type(2))) float v2f;
typedef __attribute__((ext_vector_type(8))) float v8f;

#define DIN  128
#define DOUT 64

// ---------------------------------------------------------------------------
// Kernel 1: support = x @ W via V_WMMA_F32_16X16X4_F32 (CDNA5 fp32 matrix op)
// Block = 256 threads = 8 wave32; each wave computes a 16x64 output tile.
// W staged in LDS in WMMA-native interleaved layout:
//   ldsW2[(k>>1)*2*DOUT + n*2 + (k&1)] = W[k][n]
// so each B fragment {W[k][n], W[k+1][n]} is one aligned 8-byte LDS read and
// the 4 N-tile fragments per K-step fuse into ds_load_2addr_b64 pairs.
// ---------------------------------------------------------------------------
__global__ __launch_bounds__(256) void gemm_wmma(const float* __restrict__ x,
                                                 const float* __restrict__ w,
                                                 float* __restrict__ support,
                                                 int nrows) {
  __shared__ __align__(16) float ldsW2[DIN * DOUT];  // 32 KB of 320 KB WGP LDS
  const int tid = threadIdx.x;
  // cooperative staging with the K-pair interleave swizzle (one-time cost)
  for (int base = tid * 4; base < DIN * DOUT; base += 256 * 4) {
    const float4 v = *(const float4*)&w[base];
    const int k    = base / DOUT;   // row of W (K index)
    const int n    = base % DOUT;   // col of W (N index)
    const int pair = k >> 1;
    const int sub  = k & 1;
    float* dst = &ldsW2[pair * (2 * DOUT) + n * 2 + sub];
    dst[0] = v.x;
    dst[2] = v.y;
    dst[4] = v.z;
    dst[6] = v.w;
  }
  __syncthreads();

  const int wave = tid >> 5;                 // 0..7
  const int lane = tid & 31;                 // wave32 lane
  const int tile = blockIdx.x * 8 + wave;    // 16-row tile index
  const int rowBase = tile * 16;
  if (rowBase >= nrows) return;              // uniform per-wave exit: EXEC stays all-1s

  // ISA 7.12.2 layouts for f32 16x16x4:
  //  A (16x4):  lane m=lane&15 holds row M=m; lane half g=lane>>4 selects K pair
  //             VGPR0 = K(2g+0), VGPR1 = K(2g+1)
  //  B (4x16):  lane m holds col N=m; same K-pair split across lane halves
  //  C/D:       VGPR r -> M = r + 8*g, lane m -> N = m
  const int m = lane & 15;
  const int g = lane >> 4;

  v8f c0 = {}, c1 = {}, c2 = {}, c3 = {};

  const float* xrow = x + (size_t)(rowBase + m) * DIN;
  // per-lane base into the interleaved W: pair index (k0/2 + g), column m
  const float* wbase = &ldsW2[g * (2 * DOUT) + m * 2];

#pragma unroll 4
  for (int k0 = 0; k0 < DIN; k0 += 4) {
    const int k = k0 + 2 * g;
    v2f a;
    a.x = xrow[k];
    a.y = xrow[k + 1];

    // fragments for the 4 N-tiles: byte offsets 0 / 128 / 256 / 384 from base
    const float* wl = wbase + (k0 >> 1) * (2 * DOUT);
    const v2f b0 = *(const v2f*)(wl + 0);
    const v2f b1 = *(const v2f*)(wl + 32);
    const v2f b2 = *(const v2f*)(wl + 64);
    const v2f b3 = *(const v2f*)(wl + 96);

    c0 = __builtin_amdgcn_wmma_f32_16x16x4_f32(false, a, false, b0, (short)0, c0, false, false);
    c1 = __builtin_amdgcn_wmma_f32_16x16x4_f32(false, a, false, b1, (short)0, c1, false, false);
    c2 = __builtin_amdgcn_wmma_f32_16x16x4_f32(false, a, false, b2, (short)0, c2, false, false);
    c3 = __builtin_amdgcn_wmma_f32_16x16x4_f32(false, a, false, b3, (short)0, c3, false, false);
  }

  // Store D tiles: per VGPR r, lanes are contiguous in N -> coalesced b32 stores.
  // Whole-tile uniform guard: avoids per-row exec-mask churn in the epilogue.
  float* outBase = support + (size_t)rowBase * DOUT + m;
  if (rowBase + 16 <= nrows) {               // common (and, for N%16==0, only) path
#pragma unroll
    for (int r = 0; r < 8; ++r) {
      const int row = r + 8 * g;
      outBase[(size_t)row * DOUT +  0] = c0[r];
      outBase[(size_t)row * DOUT + 16] = c1[r];
      outBase[(size_t)row * DOUT + 32] = c2[r];
      outBase[(size_t)row * DOUT + 48] = c3[r];
    }
  } else {                                   // ragged tail tile
#pragma unroll
    for (int r = 0; r < 8; ++r) {
      const int row = r + 8 * g;
      if (rowBase + row < nrows) {
        outBase[(size_t)row * DOUT +  0] = c0[r];
        outBase[(size_t)row * DOUT + 16] = c1[r];
        outBase[(size_t)row * DOUT + 32] = c2[r];
        outBase[(size_t)row * DOUT + 48] = c3[r];
      }
    }
  }
}

// ---------------------------------------------------------------------------
// Kernel 2: out[n][d] = bias[d]  (baseline for the scatter-add)
// ---------------------------------------------------------------------------
__global__ void init_out(float* __restrict__ out, const float* __restrict__ bias,
                         int n) {
  const int i = blockIdx.x * blockDim.x + threadIdx.x;
  if (i < n) out[i] = bias[i & (DOUT - 1)];
}

// ---------------------------------------------------------------------------
// Kernel 3: out[row] += ev[e] * support[col]   (16 lanes per edge, float4 path)
// support/out are L2-resident (25.6 MB each, 192 MB L2); hardware f32 atomics.
// ---------------------------------------------------------------------------
__global__ __launch_bounds__(256) void spmm_scatter(const int* __restrict__ ei,
                                                    const float* __restrict__ ev,
                                                    const float* __restrict__ support,
                                                    float* __restrict__ out,
                                                    int E) {
  const int t = blockIdx.x * blockDim.x + threadIdx.x;
  const int e = t >> 4;
  if (e >= E) return;
  const int d   = (t & 15) * 4;
  const int row = ei[e];        // edge_index[0][e]
  const int col = ei[E + e];    // edge_index[1][e]
  const float v = ev[e];

  const float4 s = *(const float4*)&support[(size_t)col * DOUT + d];  // global_load_b128
  float* o = &out[(size_t)row * DOUT + d];
  unsafeAtomicAdd(o + 0, v * s.x);   // global_atomic_add_f32 (no return)
  unsafeAtomicAdd(o + 1, v * s.y);
  unsafeAtomicAdd(o + 2, v * s.z);
  unsafeAtomicAdd(o + 3, v * s.w);
}

// ---------------------------------------------------------------------------
extern "C" void kernel_launch(void* const* d_in, const int* in_sizes, int n_in,
                              void* d_out, int out_size, void* d_ws, size_t ws_size,
                              hipStream_t stream) {
  const float* x    = (const float*)d_in[0];  // [N, 128]
  const int*   ei   = (const int*)  d_in[1];  // [2, E]
  const float* ev   = (const float*)d_in[2];  // [E]
  const float* w    = (const float*)d_in[3];  // [128, 64]
  const float* bias = (const float*)d_in[4];  // [64]
  float* out = (float*)d_out;                 // [N, 64]

  const int N = in_sizes[0] / DIN;
  const int E = in_sizes[2];

  float* support = (float*)d_ws;              // N*64 f32 = 25.6 MB scratch

  // Phase 1: dense projection on the f32 WMMA pipe
  const int tiles = (N + 15) / 16;
  const int gemmBlocks = (tiles + 7) / 8;
  gemm_wmma<<<gemmBlocks, 256, 0, stream>>>(x, w, support, N);

  // Phase 2: seed output with bias
  const int outElems = N * DOUT;
  init_out<<<(outElems + 255) / 256, 256, 0, stream>>>(out, bias, outElems);

  // Phase 3: edge scatter with hardware f32 atomics
  const long long threads = (long long)E * 16;
  spmm_scatter<<<(int)((threads + 255) / 256), 256, 0, stream>>>(ei, ev, support, out, E);
}